// DKM_3367254360577
// MI455X (gfx1250) — compile-verified
//
#include <hip/hip_runtime.h>

// DKM (soft k-means) for MI455X / gfx1250, fp32 via V_WMMA_F32_16X16X4_F32.
// N=32768 points, D=512 dims, K=256 centroids, up to 26 steps.
// d_out = [ C (256*512) | a (32768*256) ] fp32.

typedef __attribute__((ext_vector_type(2))) float v2f;
typedef __attribute__((ext_vector_type(8))) float v8f;

static constexpr int   NPTS   = 32768;
static constexpr int   DIMS   = 512;
static constexpr int   KCL    = 256;
static constexpr float TEMPI  = 2.0f;    // 1 / TEMP, TEMP = 0.5
static constexpr float THRESH = 1e-4f;
static constexpr float EPSV   = 1e-6f;

__device__ __forceinline__ v8f wmma_f32(v2f a, v2f b, v8f c) {
  // (neg_a, A, neg_b, B, c_mod, C, reuse_a, reuse_b)
  return __builtin_amdgcn_wmma_f32_16x16x4_f32(false, a, false, b, (short)0, c,
                                               false, false);
}

// ---- per-row squared norms (X once; C each iteration). 8 waves -> 8 rows/block
__global__ __launch_bounds__(256) void rowsq_kernel(const float* __restrict__ M,
                                                    float* __restrict__ out,
                                                    int nrows,
                                                    const int* __restrict__ conv) {
  if (conv && *conv) return;
  const int wave = threadIdx.x >> 5;
  const int lane = threadIdx.x & 31;
  const int row  = blockIdx.x * 8 + wave;
  if (row >= nrows) return;
  const float* p = M + (size_t)row * DIMS;
  float s = 0.f;
  for (int j = lane; j < DIMS; j += 32) { float v = p[j]; s += v * v; }
  #pragma unroll
  for (int off = 16; off > 0; off >>= 1) s += __shfl_xor(s, off, 32);
  if (lane == 0) out[row] = s;
}

// ---- init: C_cur = C_init, converged = 0
__global__ __launch_bounds__(256) void init_kernel(float* __restrict__ Ccur,
                                                   const float* __restrict__ Cinit,
                                                   int* __restrict__ conv) {
  int i = blockIdx.x * 256 + threadIdx.x;
  if (i < KCL * DIMS) Ccur[i] = Cinit[i];
  if (i == 0) *conv = 0;
}

// ---- zero accumulators for this iteration
__global__ __launch_bounds__(256) void zero_kernel(float* __restrict__ Cacc,
                                                   float* __restrict__ colsum,
                                                   float* __restrict__ diff) {
  int i = blockIdx.x * 256 + threadIdx.x;
  if (i < KCL * DIMS) Cacc[i] = 0.f;
  if (i < KCL) colsum[i] = 0.f;
  if (i == 0) *diff = 0.f;
}

// ---- assignment step: logits = -2*sqrt(max(x2 + c2 - 2 X.C, 1e-12)),
//      row softmax over K=256, a -> global, colsum += column sums.
// Block: 32 rows x 256 centroids. 8 waves, each a 32x32 WMMA tile, K-dim = 512.
__global__ __launch_bounds__(256) void assign_kernel(
    const float* __restrict__ X, const float* __restrict__ Ccur,
    const float* __restrict__ x2, const float* __restrict__ c2,
    float* __restrict__ Aout, float* __restrict__ colsum,
    const int* __restrict__ conv) {
  if (*conv) return;

  __shared__ float lds[32][260];   // logits tile, padded (stride 260 -> low bank conflict)
  __shared__ float rowmax[32];
  __shared__ float rowinv[32];

  const int tid    = threadIdx.x;
  const int wave   = tid >> 5;     // 0..7, splits the N (centroid) direction
  const int lane   = tid & 31;
  const int laneLo = lane & 15;
  const int laneHi = lane >> 4;
  const int rowBase = blockIdx.x * 32;
  const int colBase = wave * 32;

  // A frag (16x4 f32): lane<16 holds {K=0,K=1} for M=lane; lane>=16 {K=2,K=3}
  const float* Xw = X    + (size_t)(rowBase + laneLo) * DIMS + 2 * laneHi;
  const float* Cw = Ccur + (size_t)(colBase + laneLo) * DIMS + 2 * laneHi;

  v8f acc[2][2];
  #pragma unroll
  for (int i = 0; i < 2; ++i)
    #pragma unroll
    for (int j = 0; j < 2; ++j) acc[i][j] = (v8f)0.0f;

  for (int kb = 0; kb < DIMS; kb += 4) {
    v2f afr[2], bfr[2];
    #pragma unroll
    for (int mi = 0; mi < 2; ++mi)
      afr[mi] = *(const v2f*)(Xw + (size_t)(16 * mi) * DIMS + kb);
    #pragma unroll
    for (int ni = 0; ni < 2; ++ni)
      bfr[ni] = *(const v2f*)(Cw + (size_t)(16 * ni) * DIMS + kb);
    #pragma unroll
    for (int mi = 0; mi < 2; ++mi)
      #pragma unroll
      for (int ni = 0; ni < 2; ++ni)
        acc[mi][ni] = wmma_f32(afr[mi], bfr[ni], acc[mi][ni]);
  }

  // C/D layout: VGPR v, lanes 0-15 -> M=v, N=lane; lanes 16-31 -> M=v+8
  #pragma unroll
  for (int mi = 0; mi < 2; ++mi) {
    #pragma unroll
    for (int ni = 0; ni < 2; ++ni) {
      #pragma unroll
      for (int v = 0; v < 8; ++v) {
        int lr = mi * 16 + v + 8 * laneHi;          // 0..31 local row
        int lc = colBase + ni * 16 + laneLo;        // 0..255 centroid
        float dot = acc[mi][ni][v];
        float d2  = fmaxf(x2[rowBase + lr] + c2[lc] - 2.0f * dot, 1e-12f);
        lds[lr][lc] = -sqrtf(d2) * TEMPI;           // logit = -d / TEMP
      }
    }
  }
  __syncthreads();

  if (tid < 32) {
    float m = -3.4e38f;
    for (int k = 0; k < KCL; ++k) m = fmaxf(m, lds[tid][k]);
    float s = 0.f;
    for (int k = 0; k < KCL; ++k) s += __expf(lds[tid][k] - m);
    rowmax[tid] = m;
    rowinv[tid] = 1.0f / s;
  }
  __syncthreads();

  {   // one column per thread: write a (coalesced) + column partial sums
    int k = tid;
    float csum = 0.f;
    for (int r = 0; r < 32; ++r) {
      float av = __expf(lds[r][k] - rowmax[r]) * rowinv[r];
      Aout[(size_t)(rowBase + r) * KCL + k] = av;
      csum += av;
    }
    atomicAdd(&colsum[k], csum);
  }
}

// ---- centroid update: Cacc += a^T @ X  (reduction dim N split into 16 chunks)
// Block tile 64(K) x 128(D); 8 waves (2x4), wave tile 32x32.
__global__ __launch_bounds__(256) void update_kernel(
    const float* __restrict__ X, const float* __restrict__ Aout,
    float* __restrict__ Cacc, const int* __restrict__ conv) {
  if (*conv) return;
  const int tid    = threadIdx.x;
  const int wave   = tid >> 5;
  const int lane   = tid & 31;
  const int laneLo = lane & 15;
  const int laneHi = lane >> 4;
  const int wm = wave >> 2;                         // 0..1
  const int wn = wave & 3;                          // 0..3
  const int m0 = blockIdx.x * 64 + wm * 32;         // centroid base
  const int n0 = blockIdx.y * 128 + wn * 32;        // dim base
  const int rbase = blockIdx.z * (NPTS / 16);       // chunk of points

  v8f acc[2][2];
  #pragma unroll
  for (int i = 0; i < 2; ++i)
    #pragma unroll
    for (int j = 0; j < 2; ++j) acc[i][j] = (v8f)0.0f;

  for (int r = rbase; r < rbase + NPTS / 16; r += 4) {
    const int r0 = r + 2 * laneHi;
    v2f afr[2], bfr[2];
    // A = a^T: A[m][k] = a[r+k][m]; coalesced across lanes (consecutive m)
    #pragma unroll
    for (int mi = 0; mi < 2; ++mi) {
      afr[mi].x = Aout[(size_t)r0 * KCL + m0 + 16 * mi + laneLo];
      afr[mi].y = Aout[(size_t)(r0 + 1) * KCL + m0 + 16 * mi + laneLo];
    }
    #pragma unroll
    for (int ni = 0; ni < 2; ++ni) {
      bfr[ni].x = X[(size_t)r0 * DIMS + n0 + 16 * ni + laneLo];
      bfr[ni].y = X[(size_t)(r0 + 1) * DIMS + n0 + 16 * ni + laneLo];
    }
    #pragma unroll
    for (int mi = 0; mi < 2; ++mi)
      #pragma unroll
      for (int ni = 0; ni < 2; ++ni)
        acc[mi][ni] = wmma_f32(afr[mi], bfr[ni], acc[mi][ni]);
  }

  #pragma unroll
  for (int mi = 0; mi < 2; ++mi)
    #pragma unroll
    for (int ni = 0; ni < 2; ++ni)
      #pragma unroll
      for (int v = 0; v < 8; ++v) {
        int m = m0 + 16 * mi + v + 8 * laneHi;
        int d = n0 + 16 * ni + laneLo;
        atomicAdd(&Cacc[(size_t)m * DIMS + d], acc[mi][ni][v]);
      }
}

// ---- normalize C_new and accumulate L1 diff vs current C
__global__ __launch_bounds__(256) void finalize_kernel(
    float* __restrict__ Cacc, const float* __restrict__ Ccur,
    const float* __restrict__ colsum, float* __restrict__ diff,
    const int* __restrict__ conv) {
  if (*conv) return;
  __shared__ float red[256];
  const int idx = blockIdx.x * 256 + threadIdx.x;
  const int k = idx >> 9;                           // / DIMS
  float v = Cacc[idx] / (colsum[k] + EPSV);
  Cacc[idx] = v;
  red[threadIdx.x] = fabsf(v - Ccur[idx]);
  __syncthreads();
  #pragma unroll
  for (int s = 128; s > 0; s >>= 1) {
    if (threadIdx.x < s) red[threadIdx.x] += red[threadIdx.x + s];
    __syncthreads();
  }
  if (threadIdx.x == 0) atomicAdd(diff, red[0]);
}

// ---- lax.cond: if diff <= THRESH freeze (keep C,a of this step), else C<-C_new
__global__ __launch_bounds__(256) void commit_kernel(
    float* __restrict__ Ccur, const float* __restrict__ Cacc,
    const float* __restrict__ diff, int* __restrict__ conv) {
  if (*conv) return;
  const int idx = blockIdx.x * 256 + threadIdx.x;
  if (*diff <= THRESH) {
    if (idx == 0) *conv = 1;
    return;
  }
  Ccur[idx] = Cacc[idx];
}

__global__ __launch_bounds__(256) void copy_kernel(float* __restrict__ dst,
                                                   const float* __restrict__ src,
                                                   int n) {
  int i = blockIdx.x * 256 + threadIdx.x;
  if (i < n) dst[i] = src[i];
}

extern "C" void kernel_launch(void* const* d_in, const int* in_sizes, int n_in,
                              void* d_out, int out_size, void* d_ws, size_t ws_size,
                              hipStream_t stream) {
  const float* X     = (const float*)d_in[0];   // [32768, 512]
  const float* Cinit = (const float*)d_in[1];   // [256, 512]
  float* out  = (float*)d_out;                  // C then a
  float* aout = out + KCL * DIMS;               // [32768, 256]

  float* ws     = (float*)d_ws;
  float* Ccur   = ws;                    // 131072
  float* Cacc   = ws + 131072;           // 131072
  float* x2     = ws + 262144;           // 32768
  float* c2     = ws + 294912;           // 256
  float* colsum = ws + 295168;           // 256
  float* diff   = ws + 295424;           // 1
  int*   conv   = (int*)(ws + 295425);   // 1

  init_kernel<<<512, 256, 0, stream>>>(Ccur, Cinit, conv);
  rowsq_kernel<<<NPTS / 8, 256, 0, stream>>>(X, x2, NPTS, nullptr);

  for (int t = 0; t < 26; ++t) {         // 1 initial step + 25 scan iterations
    rowsq_kernel<<<KCL / 8, 256, 0, stream>>>(Ccur, c2, KCL, conv);
    zero_kernel<<<512, 256, 0, stream>>>(Cacc, colsum, diff);
    assign_kernel<<<NPTS / 32, 256, 0, stream>>>(X, Ccur, x2, c2, aout, colsum, conv);
    update_kernel<<<dim3(KCL / 64, DIMS / 128, 16), 256, 0, stream>>>(X, aout, Cacc, conv);
    finalize_kernel<<<512, 256, 0, stream>>>(Cacc, Ccur, colsum, diff, conv);
    if (t < 25) commit_kernel<<<512, 256, 0, stream>>>(Ccur, Cacc, diff, conv);
  }
  copy_kernel<<<512, 256, 0, stream>>>(out, Ccur, KCL * DIMS);
}